// BarycentricCoordinates_2199023256226
// MI455X (gfx1250) — compile-verified
//
#include <hip/hip_runtime.h>
#include <math.h>

typedef float v2f __attribute__((ext_vector_type(2)));
typedef float v8f __attribute__((ext_vector_type(8)));

#define KNN    32
#define TILE   16
#define RADIUS_F 0.15f
#define TPL_RADIUS_F (0.75f * 0.15f)
#define N_RAD  5
#define N_TPL  40

// ---------------------------------------------------------------------------
// Stage 1: KNN via WMMA Gram tiles.
// One wave owns 16 query vertices. Per iteration it processes 32 candidates
// with two V_WMMA_F32_16X16X4_F32 ops. The 4th K slot carries the candidate
// norm with A pre-scaled by -2, so the accumulator is directly the per-row
// selection key  |c|^2 - 2 q.c  (the row-constant |q|^2 cannot change the
// per-query top-K order, so it is dropped). A per-query threshold + wave
// ballot skips the serial insertion phase when no candidate can qualify.
// ---------------------------------------------------------------------------
__global__ __launch_bounds__(32) void bc_knn_wmma_kernel(
    const float* __restrict__ verts, int* __restrict__ nidx, int V) {
  const int lane = threadIdx.x;
  const int tilesPerBatch = V / TILE;
  const int b  = blockIdx.x / tilesPerBatch;
  const int q0 = (blockIdx.x % tilesPerBatch) * TILE;
  const float* vb = verts + (size_t)b * V * 3;

  __shared__ float s_d2[TILE][2 * TILE + 1];
  __shared__ float s_top_d[TILE][KNN];
  __shared__ int   s_top_i[TILE][KNN];
  __shared__ float s_worst[TILE];

  // A layout (16x4): lanes 0-15 hold K=0,1; lanes 16-31 hold K=2,3.
  // Row m = (-2qx, -2qy, -2qz, 1).
  const int m = lane & 15;
  const float qx = vb[(q0 + m) * 3 + 0];
  const float qy = vb[(q0 + m) * 3 + 1];
  const float qz = vb[(q0 + m) * 3 + 2];
  v2f amat;
  if (lane < 16) { amat.x = -2.0f * qx; amat.y = -2.0f * qy; }
  else           { amat.x = -2.0f * qz; amat.y = 1.0f; }

  for (int q = 0; q < TILE; ++q) { s_top_d[q][lane] = 3.0e38f; s_top_i[q][lane] = 0; }
  if (lane < TILE) s_worst[lane] = 3.0e38f;
  __syncthreads();

  const int hi = lane >> 4;
  const int n  = lane & 15;
  for (int c0 = 0; c0 < V; c0 += 2 * TILE) {
    // Two candidate tiles; every lane loads full xyz of its column.
    const float a0x = vb[(c0 + n) * 3 + 0];
    const float a0y = vb[(c0 + n) * 3 + 1];
    const float a0z = vb[(c0 + n) * 3 + 2];
    const float a1x = vb[(c0 + TILE + n) * 3 + 0];
    const float a1y = vb[(c0 + TILE + n) * 3 + 1];
    const float a1z = vb[(c0 + TILE + n) * 3 + 2];
    if (c0 + 4 * TILE <= V)  // prefetch next iteration's tiles
      __builtin_prefetch(&vb[(c0 + 2 * TILE + n) * 3], 0, 1);

    // B layout (4x16): col n = (cx, cy, cz, |c|^2).
    v2f bm0, bm1;
    if (lane < 16) {
      bm0.x = a0x; bm0.y = a0y;
      bm1.x = a1x; bm1.y = a1y;
    } else {
      bm0.x = a0z; bm0.y = a0x * a0x + a0y * a0y + a0z * a0z;
      bm1.x = a1z; bm1.y = a1x * a1x + a1y * a1y + a1z * a1z;
    }

    v8f acc0 = {}, acc1 = {};
    acc0 = __builtin_amdgcn_wmma_f32_16x16x4_f32(
        false, amat, false, bm0, (short)0, acc0, false, false);
    acc1 = __builtin_amdgcn_wmma_f32_16x16x4_f32(
        false, amat, false, bm1, (short)0, acc1, false, false);

    // C/D layout: element r is (M = r + 8*(lane>>4), N = lane&15).
    bool anyf = false;
    #pragma unroll
    for (int r = 0; r < 8; ++r) {
      const int M = r + 8 * hi;
      const float w = s_worst[M];
      s_d2[M][n]        = acc0[r];
      s_d2[M][TILE + n] = acc1[r];
      anyf |= (acc0[r] < w) | (acc1[r] < w);
    }
    __syncthreads();

    if (__ballot(anyf) != 0ull) {
      // Lane q < 16 owns query q: insertion into sorted top-32 list.
      if (lane < TILE) {
        float worst = s_top_d[lane][KNN - 1];
        for (int j = 0; j < 2 * TILE; ++j) {
          const float d = s_d2[lane][j];
          if (d < worst) {
            int pos = KNN - 1;
            while (pos > 0 && s_top_d[lane][pos - 1] > d) {
              s_top_d[lane][pos] = s_top_d[lane][pos - 1];
              s_top_i[lane][pos] = s_top_i[lane][pos - 1];
              --pos;
            }
            s_top_d[lane][pos] = d;
            s_top_i[lane][pos] = c0 + j;
            worst = s_top_d[lane][KNN - 1];
          }
        }
        s_worst[lane] = worst;
      }
      __syncthreads();
    }
  }

  if (lane < TILE) {
    const size_t base = ((size_t)b * V + q0 + lane) * KNN;
    for (int k = 0; k < KNN; ++k) nidx[base + k] = s_top_i[lane][k];
  }
}

// ---------------------------------------------------------------------------
// Stage 2 helpers
// ---------------------------------------------------------------------------
__device__ __forceinline__ float wave_sum32(float x) {
  #pragma unroll
  for (int off = 16; off > 0; off >>= 1) x += __shfl_xor(x, off, 32);
  return x;
}

__device__ __forceinline__ void wave_argmin32(float v, int i, float& mv, int& mi) {
  mv = v; mi = i;
  #pragma unroll
  for (int off = 16; off > 0; off >>= 1) {
    const float ov = __shfl_xor(mv, off, 32);
    const int   oi = __shfl_xor(mi, off, 32);
    if (ov < mv || (ov == mv && oi < mi)) { mv = ov; mi = oi; }
  }
}

__device__ __forceinline__ void cross3(const float* a, const float* c, float* o) {
  o[0] = a[1] * c[2] - a[2] * c[1];
  o[1] = a[2] * c[0] - a[0] * c[2];
  o[2] = a[0] * c[1] - a[1] * c[0];
}

__device__ __forceinline__ void eigvec3(float a00, float a01, float a02,
                                        float a11, float a12, float a22,
                                        float lam, float* v) {
  const float r0[3] = {a00 - lam, a01, a02};
  const float r1[3] = {a01, a11 - lam, a12};
  const float r2[3] = {a02, a12, a22 - lam};
  float c01[3], c02[3], c12[3];
  cross3(r0, r1, c01); cross3(r0, r2, c02); cross3(r1, r2, c12);
  const float n01 = c01[0]*c01[0] + c01[1]*c01[1] + c01[2]*c01[2];
  const float n02 = c02[0]*c02[0] + c02[1]*c02[1] + c02[2]*c02[2];
  const float n12 = c12[0]*c12[0] + c12[1]*c12[1] + c12[2]*c12[2];
  const float* best = c01; float nb = n01;
  if (n02 > nb) { best = c02; nb = n02; }
  if (n12 > nb) { best = c12; nb = n12; }
  const float inv = rsqrtf(fmaxf(nb, 1e-30f));
  v[0] = best[0] * inv; v[1] = best[1] * inv; v[2] = best[2] * inv;
}

// ---------------------------------------------------------------------------
// Stage 2: one wave per vertex; lane k owns neighbor k (K == wave32 == 32).
// ---------------------------------------------------------------------------
__global__ __launch_bounds__(128) void bc_frame_kernel(
    const float* __restrict__ verts, const int* __restrict__ nidx,
    float* __restrict__ out, int V, int total) {
  const int lane = threadIdx.x & 31;
  const int gv = blockIdx.x * 4 + (threadIdx.x >> 5);
  if (gv >= total) return;
  const int b = gv / V;
  const int v = gv % V;
  const float* vb = verts + (size_t)b * V * 3;

  const float px0 = vb[v * 3 + 0], py0 = vb[v * 3 + 1], pz0 = vb[v * 3 + 2];
  const int ni = nidx[((size_t)b * V + v) * KNN + lane];
  const float nx = vb[ni * 3 + 0] - px0;
  const float ny = vb[ni * 3 + 1] - py0;
  const float nz = vb[ni * 3 + 2] - pz0;

  // Weighted covariance
  const float dlen = sqrtf(nx * nx + ny * ny + nz * nz);
  const float w = fmaxf(RADIUS_F - dlen, 0.0f);
  const float wtot = wave_sum32(w) + 1e-8f;
  const float cxx = wave_sum32(w * nx * nx) / wtot;
  const float cxy = wave_sum32(w * nx * ny) / wtot;
  const float cxz = wave_sum32(w * nx * nz) / wtot;
  const float cyy = wave_sum32(w * ny * ny) / wtot;
  const float cyz = wave_sum32(w * ny * nz) / wtot;
  const float czz = wave_sum32(w * nz * nz) / wtot;

  // 3x3 symmetric eigensolver (trig method); z = min evec, x = max evec.
  float zvec[3], xvec[3];
  const float qd = (cxx + cyy + czz) * (1.0f / 3.0f);
  const float p1 = cxy * cxy + cxz * cxz + cyz * cyz;
  const float p2 = (cxx - qd) * (cxx - qd) + (cyy - qd) * (cyy - qd) +
                   (czz - qd) * (czz - qd) + 2.0f * p1;
  if (p2 < 1e-24f) {
    zvec[0] = 1.0f; zvec[1] = 0.0f; zvec[2] = 0.0f;
    xvec[0] = 0.0f; xvec[1] = 0.0f; xvec[2] = 1.0f;
  } else {
    const float p = sqrtf(p2 * (1.0f / 6.0f));
    const float ip = 1.0f / p;
    const float b00 = (cxx - qd) * ip, b01 = cxy * ip, b02 = cxz * ip;
    const float b11 = (cyy - qd) * ip, b12 = cyz * ip, b22 = (czz - qd) * ip;
    const float detB = b00 * (b11 * b22 - b12 * b12)
                     - b01 * (b01 * b22 - b12 * b02)
                     + b02 * (b01 * b12 - b11 * b02);
    const float r = fminf(fmaxf(detB * 0.5f, -1.0f), 1.0f);
    const float phi = acosf(r) * (1.0f / 3.0f);
    const float emax = qd + 2.0f * p * cosf(phi);
    const float emin = qd + 2.0f * p * cosf(phi + 2.0943951023931953f);
    eigvec3(cxx, cxy, cxz, cyy, cyz, czz, emin, zvec);
    eigvec3(cxx, cxy, cxz, cyy, cyz, czz, emax, xvec);
  }

  // Sign disambiguation (matches reference: sign(sum(neigh . axis) + 1e-12))
  {
    const float sx = wave_sum32(nx * xvec[0] + ny * xvec[1] + nz * xvec[2]) + 1e-12f;
    const float fx = (sx >= 0.0f) ? 1.0f : -1.0f;
    xvec[0] *= fx; xvec[1] *= fx; xvec[2] *= fx;
    const float sz = wave_sum32(nx * zvec[0] + ny * zvec[1] + nz * zvec[2]) + 1e-12f;
    const float fz = (sz >= 0.0f) ? 1.0f : -1.0f;
    zvec[0] *= fz; zvec[1] *= fz; zvec[2] *= fz;
  }
  float yvec[3];
  cross3(zvec, xvec, yvec);

  // Log map: project out normal (z), express in (x, y), rescale to |neigh|.
  const float ndz = nx * zvec[0] + ny * zvec[1] + nz * zvec[2];
  const float prx = nx - ndz * zvec[0];
  const float pry = ny - ndz * zvec[1];
  const float prz = nz - ndz * zvec[2];
  const float px3 = prx * xvec[0] + pry * xvec[1] + prz * xvec[2];
  const float py3 = prx * yvec[0] + pry * yvec[1] + prz * yvec[2];
  const float invn = rsqrtf(fmaxf(px3 * px3 + py3 * py3, 1e-12f));
  const float Px = dlen * px3 * invn;
  const float Py = dlen * py3 * invn;

  float* outv = out + ((size_t)b * V + v) * (N_TPL * 3 * 2);
  const float fNi = (float)ni;

  for (int t = 0; t < N_TPL; ++t) {
    const int ri = t >> 3, ai = t & 7;
    const float tr = TPL_RADIUS_F * (float)(ri + 1) * (1.0f / (float)N_RAD);
    const float th = (float)ai * 0.78539816339744831f;  // 2*pi/8
    const float tx = tr * cosf(th);
    const float ty = tr * sinf(th);

    const float dx = tx - Px, dy = ty - Py;
    float cur = dx * dx + dy * dy;  // monotone in the reference's norm

    int c0, c1, c2;
    {
      float mv; int mi;
      wave_argmin32(cur, lane, mv, mi);
      c0 = mi; if (lane == mi) cur = 3.0e38f;
      wave_argmin32(cur, lane, mv, mi);
      c1 = mi; if (lane == mi) cur = 3.0e38f;
      wave_argmin32(cur, lane, mv, mi);
      c2 = mi;
    }

    const float p0x = __shfl(Px, c0), p0y = __shfl(Py, c0);
    const float p1x = __shfl(Px, c1), p1y = __shfl(Py, c1);
    const float p2x = __shfl(Px, c2), p2y = __shfl(Py, c2);
    const float i0 = __shfl(fNi, c0);
    const float i1 = __shfl(fNi, c1);
    const float i2 = __shfl(fNi, c2);

    // Barycentric coords vs. triangle (closest0, closest1, closest2).
    const float v0x = p2x - p0x, v0y = p2y - p0y;   // p[2] - p[0]
    const float v1x = p1x - p0x, v1y = p1y - p0y;   // p[1] - p[0]
    const float v2x = tx - p0x,  v2y = ty - p0y;    // T    - p[0]
    const float dot00 = v0x * v0x + v0y * v0y;
    const float dot01 = v0x * v1x + v0y * v1y;
    const float dot02 = v0x * v2x + v0y * v2y;
    const float dot11 = v1x * v1x + v1y * v1y;
    const float dot12 = v1x * v2x + v1y * v2y;
    const float denom = dot00 * dot11 - dot01 * dot01;
    const float w2 = (dot11 * dot02 - dot01 * dot12) / denom;
    const float w1 = (dot00 * dot12 - dot01 * dot02) / denom;
    const float w0 = 1.0f - w2 - w1;

    if (lane == (t & 31)) {
      float* o = outv + t * 6;
      o[0] = i0; o[1] = w2;   // stack([pidx, weights]) with weights=[w2,w1,w0]
      o[2] = i1; o[3] = w1;
      o[4] = i2; o[5] = w0;
    }
  }
}

// ---------------------------------------------------------------------------
extern "C" void kernel_launch(void* const* d_in, const int* in_sizes, int n_in,
                              void* d_out, int out_size, void* d_ws, size_t ws_size,
                              hipStream_t stream) {
  (void)n_in; (void)out_size; (void)ws_size;
  const float* verts = (const float*)d_in[0];
  const int B = 2;
  const int V = in_sizes[0] / (B * 3);   // 6144
  int* nidx = (int*)d_ws;                // B*V*KNN ints = 1.5 MB

  dim3 g1(B * (V / TILE)), b1(32);
  bc_knn_wmma_kernel<<<g1, b1, 0, stream>>>(verts, nidx, V);

  const int total = B * V;
  dim3 g2((total + 3) / 4), b2(128);
  bc_frame_kernel<<<g2, b2, 0, stream>>>(verts, nidx, (float*)d_out, V, total);
}